// Aimv2VisionEmbeddings_54296976556228
// MI455X (gfx1250) — compile-verified
//
#include <hip/hip_runtime.h>
#include <math.h>

typedef __attribute__((ext_vector_type(2))) float v2f;
typedef __attribute__((ext_vector_type(8))) float v8f;

#define KDIM 588
#define KC   28                 // K-chunk staged in LDS (21 chunks)
#define HDIM 1024
#define NSEQ 4096
#define ROWS_PER_BLOCK 16
#define THREADS 256
#define TILES_PER_WAVE 8        // 8 waves * 8 tiles * 16 cols = 1024 cols

// CDNA5 async global->LDS copy (ASYNCcnt-tracked, no VGPR round trip).
// lds_off: LDS byte address (low 32 bits of the generic shared address),
// gaddr:   64-bit global address.
__device__ __forceinline__ void async_copy_b128(unsigned lds_off, const void* gaddr) {
    asm volatile("global_load_async_to_lds_b128 %0, %1, off"
                 :: "v"(lds_off), "v"((unsigned long long)(uintptr_t)gaddr)
                 : "memory");
}
__device__ __forceinline__ void wait_async0() {
    asm volatile("s_wait_asynccnt 0x0" ::: "memory");
}

__global__ __launch_bounds__(THREADS)
void aimv2_embed_kernel(const float* __restrict__ X,
                        const int*   __restrict__ shapes,
                        const float* __restrict__ W,
                        const float* __restrict__ bias,
                        const float* __restrict__ gamma,
                        float* __restrict__ out)
{
    __shared__ float lds_a[ROWS_PER_BLOCK * KDIM];  // 16 x 588 activations (37.6 KB)
    __shared__ float lds_w[HDIM * KC];              // 1024 x 28 weight chunk (114.7 KB)
    __shared__ float lds_red[ROWS_PER_BLOCK];       // per-row sum of squares

    const int tid  = threadIdx.x;
    const int wave = tid >> 5;
    const int lane = tid & 31;
    const int base = blockIdx.x * ROWS_PER_BLOCK;   // global output-row base

    // ---- stage the 16x588 activation slab into LDS (coalesced b32) ----
    {
        const float* src = X + (size_t)base * KDIM;
        for (int idx = tid; idx < ROWS_PER_BLOCK * KDIM; idx += THREADS)
            lds_a[idx] = src[idx];
    }
    if (tid < ROWS_PER_BLOCK) lds_red[tid] = 0.0f;

    const int l15  = lane & 15;                     // M (A) / col-in-tile (B,C,D)
    const int half = lane >> 4;                     // K-pair select / row-half (C,D)
    const int n0   = wave * (TILES_PER_WAVE * 16);  // this wave's first column

    // LDS fragment base pointers (all subsequent accesses use immediate offsets)
    const float* lds_arow  = &lds_a[l15 * KDIM + 2 * half];
    const float* lds_wfrag = &lds_w[(n0 + l15) * KC + 2 * half];
    const unsigned ldsw_off = (unsigned)(size_t)(void*)&lds_w[0];

    // Per-thread W staging bases: thread covers rows {tid, tid+256, tid+512, tid+768}
    const float* wblk[4];
#pragma unroll
    for (int blkI = 0; blkI < 4; ++blkI)
        wblk[blkI] = W + (size_t)(blkI * 256 + tid) * KDIM;

    v8f acc[TILES_PER_WAVE];
#pragma unroll
    for (int t = 0; t < TILES_PER_WAVE; ++t) {
        v8f z = {0.f, 0.f, 0.f, 0.f, 0.f, 0.f, 0.f, 0.f};
        acc[t] = z;
    }

    // ---- K loop: 21 chunks of (async-stage 1024x28 W tile -> 56 WMMAs) ----
    for (int c = 0; c < KDIM / KC; ++c) {
        const int k0 = c * KC;

        // async global->LDS: 28 x b128 per thread = 114.7 KB per workgroup
#pragma unroll
        for (int q = 0; q < KC / 4; ++q) {
#pragma unroll
            for (int blkI = 0; blkI < 4; ++blkI) {
                const int row = blkI * 256 + tid;
                async_copy_b128(ldsw_off + (unsigned)(row * (KC * 4) + q * 16),
                                wblk[blkI] + k0 + q * 4);
            }
        }
        wait_async0();
        __syncthreads();

        // 7 micro-steps of V_WMMA_F32_16X16X4_F32, operands fed from LDS
#pragma unroll
        for (int kk = 0; kk < KC; kk += 4) {
            v2f a = *(const v2f*)(lds_arow + k0 + kk);
            v2f bfrag[TILES_PER_WAVE];
#pragma unroll
            for (int t = 0; t < TILES_PER_WAVE; ++t)
                bfrag[t] = *(const v2f*)(lds_wfrag + t * (16 * KC) + kk);
#pragma unroll
            for (int t = 0; t < TILES_PER_WAVE; ++t)
                acc[t] = __builtin_amdgcn_wmma_f32_16x16x4_f32(
                    false, a, false, bfrag[t], (short)0, acc[t], false, false);
        }
        __syncthreads();
    }

    // ---- bias (before RMS statistics, matching reference) ----
#pragma unroll
    for (int t = 0; t < TILES_PER_WAVE; ++t) {
        float bv = bias[n0 + t * 16 + l15];
#pragma unroll
        for (int r = 0; r < 8; ++r) acc[t][r] += bv;
    }

    // ---- per-row sum of squares: shuffle across 16 lanes, ds_add_f32 across waves ----
    float p[8];
#pragma unroll
    for (int r = 0; r < 8; ++r) {
        float s = 0.0f;
#pragma unroll
        for (int t = 0; t < TILES_PER_WAVE; ++t) s += acc[t][r] * acc[t][r];
        p[r] = s;
    }
#pragma unroll
    for (int r = 0; r < 8; ++r) {
#pragma unroll
        for (int off = 1; off < 16; off <<= 1)
            p[r] += __shfl_xor(p[r], off, 32);      // stays within 16-lane half
    }
    if (l15 == 0) {
#pragma unroll
        for (int r = 0; r < 8; ++r)
            atomicAdd(&lds_red[r + 8 * half], p[r]);
    }
    __syncthreads();

    // ---- row metadata: batch, (i,j) grid position, RMS scale ----
    const int bidx  = base >> 12;                   // base / NSEQ
    const int nbase = base & (NSEQ - 1);
    const int hh = shapes[2 * bidx];
    const int ww = shapes[2 * bidx + 1];
    const int hw = hh * ww;

    float scale[8], fi[8], fj[8];
#pragma unroll
    for (int r = 0; r < 8; ++r) {
        const int m = r + 8 * half;
        scale[r] = rsqrtf(lds_red[m] * (1.0f / HDIM) + 1e-6f);
        const int n = nbase + m;
        int ii = 0, jj = 0;
        if (n < hw) { ii = n / ww; jj = n - ii * ww; }
        fi[r] = (float)ii;
        fj[r] = (float)jj;
    }

    // ---- epilogue: rms_weight * x * scale + sincos position embedding ----
    const float c_lomega = -13.287712379549449f / 256.0f;  // -log2(10000)/pos_dim
#pragma unroll
    for (int t = 0; t < TILES_PER_WAVE; ++t) {
        const int col  = n0 + t * 16 + l15;
        const float gm = gamma[col];
        const int quad = col >> 8;                  // 0:sin(j) 1:cos(j) 2:sin(i) 3:cos(i)
        const int d    = col & 255;
        const float om = exp2f((float)d * c_lomega);
#pragma unroll
        for (int r = 0; r < 8; ++r) {
            const float ang = ((quad < 2) ? fj[r] : fi[r]) * om;
            const float pos = (quad & 1) ? __cosf(ang) : __sinf(ang);
            const int m = r + 8 * half;
            out[((size_t)(base + m) << 10) + col] = acc[t][r] * scale[r] * gm + pos;
        }
    }
}

extern "C" void kernel_launch(void* const* d_in, const int* in_sizes, int n_in,
                              void* d_out, int out_size, void* d_ws, size_t ws_size,
                              hipStream_t stream) {
    const float* X      = (const float*)d_in[0];    // (B, N, 588) f32
    const int*   shapes = (const int*)  d_in[1];    // (B, 2) i32
    const float* W      = (const float*)d_in[2];    // (1024, 588) f32
    const float* bias   = (const float*)d_in[3];    // (1024,)
    const float* gamma  = (const float*)d_in[4];    // (1024,)
    float*       out    = (float*)d_out;            // (B, N, 1024) f32

    const int rows = in_sizes[0] / KDIM;            // B*N = 65536
    const int grid = rows / ROWS_PER_BLOCK;         // 4096 workgroups
    aimv2_embed_kernel<<<grid, THREADS, 0, stream>>>(X, shapes, W, bias, gamma, out);
}